// OneHotEncoder_43052752175267
// MI455X (gfx1250) — compile-verified
//
#include <hip/hip_runtime.h>
#include <stdint.h>

#define VOCAB           32000
#define TOKENS_PER_ROW  2048
#define THREADS         512
#define VEC4            (VOCAB / 4)   // 8000 float4 chunks

// One workgroup per batch row. Histogram lives entirely in LDS (128 KB),
// which only works on CDNA5's 320 KB-per-WGP LDS. Token row is streamed in
// with global_load_async_to_lds_b128; the finished histogram is streamed out
// with global_store_async_from_lds_b128 (both tracked by ASYNCcnt).
__global__ __launch_bounds__(THREADS) void onehot_hist_kernel(
    const int* __restrict__ tokens, float* __restrict__ out)
{
    __shared__ float hist[VOCAB];            // 128000 bytes
    __shared__ int   toks[TOKENS_PER_ROW];   //   8192 bytes

    const int row = blockIdx.x;
    const int tid = threadIdx.x;

    // ---- 1) async-copy this row's tokens into LDS (16 B per lane) ----
    {
        uint64_t gaddr = (uint64_t)(uintptr_t)(tokens + (size_t)row * TOKENS_PER_ROW + tid * 4);
        uint32_t laddr = (uint32_t)(uintptr_t)(&toks[tid * 4]);   // low 32 bits = LDS offset
        asm volatile("global_load_async_to_lds_b128 %0, %1, off"
                     :: "v"(laddr), "v"(gaddr) : "memory");
    }

    // ---- 2) zero the histogram while the async copy is in flight ----
    {
        float4 z; z.x = z.y = z.z = z.w = 0.0f;
        float4* h4 = (float4*)hist;
        for (int i = tid; i < VEC4; i += THREADS) h4[i] = z;      // ds_store_b128
    }

    // wait for this wave's async token load; barrier makes zeroing visible
    asm volatile("s_wait_asynccnt 0x0" ::: "memory");
    __syncthreads();

    // ---- 3) histogram via LDS float atomics (skip pad -> bin 0 stays 0) ----
    #pragma unroll
    for (int k = 0; k < 4; ++k) {
        int t = toks[tid * 4 + k];
        if (t != 0) atomicAdd(&hist[t], 1.0f);                    // ds_add_f32
    }

    __syncthreads();

    // ---- 4) drain histogram LDS -> global with async B128 stores ----
    float* orow = out + (size_t)row * VOCAB;
    for (int c = tid; c < VEC4; c += THREADS) {
        uint64_t gaddr = (uint64_t)(uintptr_t)(orow + c * 4);
        uint32_t laddr = (uint32_t)(uintptr_t)(&hist[c * 4]);
        asm volatile("global_store_async_from_lds_b128 %0, %1, off"
                     :: "v"(gaddr), "v"(laddr) : "memory");
    }
    asm volatile("s_wait_asynccnt 0x0" ::: "memory");
    // S_ENDPGM performs an implicit wait-idle as well.
}

extern "C" void kernel_launch(void* const* d_in, const int* in_sizes, int n_in,
                              void* d_out, int out_size, void* d_ws, size_t ws_size,
                              hipStream_t stream) {
    const int* tokens = (const int*)d_in[0];   // [256, 2048] int32
    // d_in[1] = lengths[256] -- unused by the reference computation.
    float* out = (float*)d_out;                // [256, 32000] f32

    const int B = in_sizes[0] / TOKENS_PER_ROW;   // 256 rows
    onehot_hist_kernel<<<B, THREADS, 0, stream>>>(tokens, out);
}